// SlideSum_23347442221550
// MI455X (gfx1250) — compile-verified
//
#include <hip/hip_runtime.h>

#define B_   64
#define L_   4096
#define F_   256
#define TL   32           // output L-rows per block
#define ROWS (TL + 2)     // staged input rows (clamped halo of 1 on each side)
#define RS   264          // LDS row stride in floats (256 + 8 pad -> conflict-free)

typedef __attribute__((ext_vector_type(2))) float v2f;
typedef __attribute__((ext_vector_type(8))) float v8f;

__global__ __launch_bounds__(256)
void SlideSum_kernel(const float* __restrict__ x,
                     const float* __restrict__ alphaPtr,
                     float* __restrict__ out) {
    __shared__ __align__(16) float smem[ROWS * RS];

    const int tid = threadIdx.x;
    const int b   = blockIdx.y;
    const int l0  = blockIdx.x * TL;

    // ---- Stage ROWS input rows into LDS via CDNA5 async global->LDS (B128) ----
    const unsigned ldsBase = (unsigned)(uintptr_t)&smem[0];
    for (int c = tid; c < ROWS * (F_ / 4); c += 256) {
        const int j  = c >> 6;     // staged row 0..ROWS-1
        const int f4 = c & 63;     // float4 column
        int g = l0 - 1 + j;        // global input row, clamped at edges
        g = g < 0 ? 0 : (g > L_ - 1 ? L_ - 1 : g);
        const unsigned long long ga =
            (unsigned long long)(uintptr_t)(x + ((size_t)b * L_ + g) * F_ + f4 * 4);
        const unsigned la = ldsBase + (unsigned)(j * RS * 4 + f4 * 16);
        asm volatile("global_load_async_to_lds_b128 %0, %1, off"
                     :: "v"(la), "v"(ga) : "memory");
    }
    const float alpha = alphaPtr[0];   // overlaps with async drain
    asm volatile("s_wait_asynccnt 0" ::: "memory");
    __syncthreads();

    const int lane = tid & 31;
    const int wave = tid >> 5;
    const int nIdx = lane & 15;   // A: row M ; B/D: column N
    const int hi   = lane >> 4;   // lane half selects K/M offset

    // 2 L-tiles x 16 F-tiles = 32 tiles per block, statically 4 per wave.
    // Constant trip count -> full unroll, no exec-masked loop (WMMA needs EXEC all-1s).
    #pragma unroll
    for (int i = 0; i < (TL / 16) * (F_ / 16) / 8; ++i) {
        const int t     = wave + 8 * i;
        const int lt    = t >> 4;
        const int f0    = (t & 15) * 16;
        const int jbase = lt * 16;          // first staged row of this tile
        const int ltl   = l0 + jbase;       // first global output row of this tile

        // band start for this lane's output row M = nIdx (shifted at grid edges)
        int s = ltl + nIdx - 1;
        s = s < 0 ? 0 : (s > L_ - 3 ? L_ - 3 : s);
        const int jsl = s - l0 + 1 - jbase;   // in [nIdx-1, nIdx+1]; band = {jsl, jsl+1, jsl+2}

        v8f acc = {};
#if __has_builtin(__builtin_amdgcn_wmma_f32_16x16x4_f32)
        #pragma unroll
        for (int kk = 0; kk < 5; ++kk) {      // K = 0..19 (18 used, 2 zero-padded)
            const int k0 = kk * 4 + 2 * hi;   // A/B VGPR0 K ; VGPR1 K = k0+1
            const int k1 = k0 + 1;
            v2f a, bm;
            a.x  = (k0 >= jsl && k0 <= jsl + 2) ? 1.0f : 0.0f;
            a.y  = (k1 >= jsl && k1 <= jsl + 2) ? 1.0f : 0.0f;
            bm.x = (k0 <= 17) ? smem[(jbase + k0) * RS + f0 + nIdx] : 0.0f;
            bm.y = (k1 <= 17) ? smem[(jbase + k1) * RS + f0 + nIdx] : 0.0f;
            acc = __builtin_amdgcn_wmma_f32_16x16x4_f32(
                false, a, false, bm, (short)0, acc, false, false);
        }
#else
        // VALU fallback: direct 3-tap from LDS
        #pragma unroll
        for (int v = 0; v < 8; ++v) {
            const int M = v + 8 * hi;
            int ss = ltl + M - 1;
            ss = ss < 0 ? 0 : (ss > L_ - 3 ? L_ - 3 : ss);
            const float* r = &smem[(ss - l0 + 1) * RS + f0 + nIdx];
            acc[v] = r[0] + r[RS] + r[2 * RS];
        }
#endif
        acc *= alpha;

        float* op = out + ((size_t)b * L_ + ltl) * F_ + f0 + nIdx;
        #pragma unroll
        for (int v = 0; v < 8; ++v) {
            const int M = v + 8 * hi;         // D layout: VGPR v -> M = v + 8*laneHalf
            op[(size_t)M * F_] = acc[v];
        }
    }
}

extern "C" void kernel_launch(void* const* d_in, const int* in_sizes, int n_in,
                              void* d_out, int out_size, void* d_ws, size_t ws_size,
                              hipStream_t stream) {
    (void)in_sizes; (void)n_in; (void)out_size; (void)d_ws; (void)ws_size;
    const float* x     = (const float*)d_in[0];
    const float* alpha = (const float*)d_in[1];
    float*       out   = (float*)d_out;
    dim3 grid(L_ / TL, B_);   // 128 L-tiles x 64 batches
    SlideSum_kernel<<<grid, 256, 0, stream>>>(x, alpha, out);
}